// MCPropNetv2_85023172591740
// MI455X (gfx1250) — compile-verified
//
#include <hip/hip_runtime.h>
#include <hip/hip_bf16.h>
#include <math.h>

typedef float v4f __attribute__((ext_vector_type(4)));

// Problem geometry (fixed by the reference)
#define BB   2
#define HH   352
#define WW   1216
#define HWP  (HH * WW)

// Tiling: 16x16 threads, 4 pixels per thread in x -> 64x16 pixel tiles.
#define BX   16
#define BY   16
#define PX   4
#define TPW  (BX * PX)        // 64 pixels wide
#define RAD  3
#define SW   (TPW + 2 * RAD)  // 70
#define SHt  (BY + 2 * RAD)   // 22
#define SWP  72               // padded LDS row stride (floats)

// CDNA5 async global->LDS path (gfx1250). Guarded so compile never breaks.
#if defined(__HIP_DEVICE_COMPILE__)
#if __has_builtin(__builtin_amdgcn_global_load_async_to_lds_b32)
#define USE_ASYNC_LDS 1
#endif
#endif

// Dk = (sum_j A_j * patch_j + A_c*(D0 - patch_c)) / (sum_j |A_j| + eps)
// for 4 consecutive pixels at once. A loads are 128-bit and non-temporal
// (284 MB/step of single-use streaming data; keep L2 for the reused tensors).
template<int K>
__device__ __forceinline__ v4f dyn_filter(const float* __restrict__ base,
                                          const float (&tile)[SHt][SWP],
                                          int ly, int lx0, v4f d0, v4f pc)
{
    v4f sa = (v4f)0.0f, sd = (v4f)0.0f, ac = (v4f)0.0f;
#pragma unroll
    for (int j = 0; j < K * K; ++j) {
        const int dy = j / K - K / 2;
        const int dx = j % K - K / 2;
        const v4f a = __builtin_nontemporal_load((const v4f*)(base + (size_t)j * HWP));
        sa.x += __builtin_fabsf(a.x);
        sa.y += __builtin_fabsf(a.y);
        sa.z += __builtin_fabsf(a.z);
        sa.w += __builtin_fabsf(a.w);
        const int r = ly + dy, c = lx0 + dx;
        sd.x = __builtin_fmaf(a.x, tile[r][c],     sd.x);
        sd.y = __builtin_fmaf(a.y, tile[r][c + 1], sd.y);
        sd.z = __builtin_fmaf(a.z, tile[r][c + 2], sd.z);
        sd.w = __builtin_fmaf(a.w, tile[r][c + 3], sd.w);
        if (j == (K * K) / 2) ac = a;
    }
    v4f D;
    D.x = (sd.x + ac.x * (d0.x - pc.x)) / (sa.x + 1e-6f);
    D.y = (sd.y + ac.y * (d0.y - pc.y)) / (sa.y + 1e-6f);
    D.z = (sd.z + ac.z * (d0.z - pc.z)) / (sa.z + 1e-6f);
    D.w = (sd.w + ac.w * (d0.w - pc.w)) / (sa.w + 1e-6f);
    return D;
}

__global__ __launch_bounds__(256) void mcprop_step(
    const float* __restrict__ Din,   // current depth estimate (B,1,H,W)
    const float* __restrict__ D0,
    const float* __restrict__ DL,
    const float* __restrict__ ML,
    const float* __restrict__ A3,    // (B,9,H,W)
    const float* __restrict__ A5,    // (B,25,H,W)
    const float* __restrict__ A7,    // (B,49,H,W)
    const float* __restrict__ SR,    // (B,3,H,W) sigma_raw
    const float* __restrict__ AL,    // alpha
    float* __restrict__ Dout)
{
    __shared__ float tile[SHt][SWP];

    const int bx = blockIdx.x, by = blockIdx.y, b = blockIdx.z;
    const int tx = threadIdx.x, ty = threadIdx.y;
    const int tid = ty * BX + tx;
    const int x0 = bx * TPW + tx * PX;
    const int y  = by * BY + ty;

    const float* Dp = Din + (size_t)b * HWP;
    const int gy0 = by * BY - RAD;
    const int gx0 = bx * TPW - RAD;

    // ---- Stage depth tile (+halo) into LDS -------------------------------
#ifdef USE_ASYNC_LDS
    const bool interior = (bx > 0) && (bx < (int)gridDim.x - 1) &&
                          (by > 0) && (by < (int)gridDim.y - 1);
    if (interior) {
        // Async DMA global -> LDS (tracked with ASYNCcnt, no VGPR staging).
        for (int i = tid; i < SHt * SW; i += BX * BY) {
            const int sy = i / SW;
            const int sx = i - sy * SW;
            const float* src = Dp + (size_t)(gy0 + sy) * WW + (gx0 + sx);
            __builtin_amdgcn_global_load_async_to_lds_b32(
                (__attribute__((address_space(1))) int*)src,
                (__attribute__((address_space(3))) int*)&tile[sy][sx],
                0, 0);
        }
        asm volatile("s_wait_asynccnt 0" ::: "memory");
    } else
#endif
    {
        // Boundary blocks: need zero padding, do it in registers.
        for (int i = tid; i < SHt * SW; i += BX * BY) {
            const int sy = i / SW;
            const int sx = i - sy * SW;
            const int gy = gy0 + sy;
            const int gx = gx0 + sx;
            float v = 0.0f;
            if (gy >= 0 && gy < HH && gx >= 0 && gx < WW)
                v = Dp[(size_t)gy * WW + gx];
            tile[sy][sx] = v;
        }
    }
    __syncthreads();

    // ---- Per-pixel dynamic filtering (4 pixels per thread) ---------------
    const int off = y * WW + x0;
    const size_t pix = (size_t)b * HWP + off;
    const int ly = ty + RAD, lx0 = tx * PX + RAD;

    const v4f d0 = *(const v4f*)(D0 + pix);
    v4f pc;
    pc.x = tile[ly][lx0];
    pc.y = tile[ly][lx0 + 1];
    pc.z = tile[ly][lx0 + 2];
    pc.w = tile[ly][lx0 + 3];

    const v4f D3 = dyn_filter<3>(A3 + (size_t)b * 9  * HWP + off, tile, ly, lx0, d0, pc);
    const v4f D5 = dyn_filter<5>(A5 + (size_t)b * 25 * HWP + off, tile, ly, lx0, d0, pc);
    const v4f D7 = dyn_filter<7>(A7 + (size_t)b * 49 * HWP + off, tile, ly, lx0, d0, pc);

    // softmax mixer + masked blend + clamp (these tensors stay L2-resident)
    const float* sr = SR + (size_t)b * 3 * HWP + off;
    const v4f s0 = *(const v4f*)sr;
    const v4f s1 = *(const v4f*)(sr + HWP);
    const v4f s2 = *(const v4f*)(sr + 2 * (size_t)HWP);
    const v4f ml = *(const v4f*)(ML + pix);
    const v4f al = *(const v4f*)(AL + pix);
    const v4f dl = *(const v4f*)(DL + pix);

    v4f res;
#pragma unroll
    for (int p = 0; p < PX; ++p) {
        const float mxv = fmaxf(s0[p], fmaxf(s1[p], s2[p]));
        const float e0 = __expf(s0[p] - mxv);
        const float e1 = __expf(s1[p] - mxv);
        const float e2 = __expf(s2[p] - mxv);
        const float mix = (e0 * D3[p] + e1 * D5[p] + e2 * D7[p]) / (e0 + e1 + e2);
        const float m  = (ml[p] > 0.5f) ? 1.0f : 0.0f;
        const float a  = fminf(fmaxf(al[p], 0.0f), 1.0f);
        const float am = a * m;
        float dt = (1.0f - am) * mix + am * dl[p];
        res[p] = fminf(fmaxf(dt, 0.0f), 80.0f);
    }
    *(v4f*)(Dout + pix) = res;
}

extern "C" void kernel_launch(void* const* d_in, const int* in_sizes, int n_in,
                              void* d_out, int out_size, void* d_ws, size_t ws_size,
                              hipStream_t stream) {
    const float* D0 = (const float*)d_in[0];
    const float* DL = (const float*)d_in[1];
    const float* ML = (const float*)d_in[2];
    const float* A3 = (const float*)d_in[3];
    const float* A5 = (const float*)d_in[4];
    const float* A7 = (const float*)d_in[5];
    const float* SR = (const float*)d_in[6];
    const float* AL = (const float*)d_in[7];
    float* out = (float*)d_out;
    float* ws0 = (float*)d_ws;   // one B*H*W float ping buffer (3.4 MB)

    dim3 grid(WW / TPW, HH / BY, BB);   // 19 x 22 x 2
    dim3 block(BX, BY);                 // 256 threads = 8 wave32 waves

    // 6 steps, ping-pong ws <-> out so the final step lands in d_out.
    const float* src = D0;
    float* bufs[2] = { ws0, out };
    for (int s = 0; s < 6; ++s) {
        float* dst = bufs[s & 1];
        mcprop_step<<<grid, block, 0, stream>>>(src, D0, DL, ML, A3, A5, A7,
                                                SR, AL, dst);
        src = dst;
    }
}